// DifferentiableGMM_78254304133662
// MI455X (gfx1250) — compile-verified
//
#include <hip/hip_runtime.h>
#include <hip/hip_bf16.h>
#include <math.h>

// GMM log-likelihood via WMMA on gfx1250.
// log_prob[n,k] = dot(feat(x_n)[8], coeff_k[8]) with
//   feat = [x0^2, x1^2, x2^2, x0, x1, x2, 1, 0]
//   coeff_k = [-.5/v0, -.5/v1, -.5/v2, m0/v0, m1/v1, m2/v2, const_k, 0]
// Two chained V_WMMA_F32_16X16X4_F32 (K=4 each) compute a 16comp x 16pt tile.

typedef __attribute__((ext_vector_type(2))) float v2f;
typedef __attribute__((ext_vector_type(8))) float v8f;

#define GMM_EPS 1e-6f
// 0.5 * D * log(2*pi), D = 3
#define HALF_D_LOG_2PI 2.7568155996140185f

__global__ __launch_bounds__(256) void gmm_logsumexp_wmma(
    const float* __restrict__ x,      // [N,3]
    const float* __restrict__ means,  // [16,3]
    const float* __restrict__ cov,    // [16,3]
    const float* __restrict__ w,      // [16]
    float* __restrict__ out,          // [N]
    int numTiles, int N)
{
  const int lane = threadIdx.x & 31;
  const int row  = lane & 15;   // A: matrix row (component); B/D: column (point)
  const int half = lane >> 4;   // 0: K=0,1 / comps 0-7 ; 1: K=2,3 / comps 8-15

  // ---- per-component coefficients (K*D is tiny; redundant compute, L2-hot) ----
  const int comp = row;
  const float v0 = cov[comp * 3 + 0] + GMM_EPS;
  const float v1 = cov[comp * 3 + 1] + GMM_EPS;
  const float v2 = cov[comp * 3 + 2] + GMM_EPS;
  const float m0 = means[comp * 3 + 0];
  const float m1 = means[comp * 3 + 1];
  const float m2 = means[comp * 3 + 2];
  const float iv0 = 1.0f / v0, iv1 = 1.0f / v1, iv2 = 1.0f / v2;
  const float constK =
      -0.5f * (m0 * m0 * iv0 + m1 * m1 * iv1 + m2 * m2 * iv2)
      - 0.5f * (__logf(v0) + __logf(v1) + __logf(v2))   // log|L| = 0.5*sum(log v)
      + __logf(w[comp]) - HALF_D_LOG_2PI;

  // A-matrix fragments (16x4 f32): lane m holds row m K=0,1; lane m+16 holds K=2,3.
  v2f a1, a2;
  if (half == 0) {
    a1 = (v2f){-0.5f * iv0, -0.5f * iv1};   // K0: x0^2, K1: x1^2
    a2 = (v2f){m1 * iv1, m2 * iv2};         // K0: x1,   K1: x2
  } else {
    a1 = (v2f){-0.5f * iv2, m0 * iv0};      // K2: x2^2, K3: x0
    a2 = (v2f){constK, 0.0f};               // K2: 1,    K3: 0 (pad)
  }

  const int wavesTotal = (int)((gridDim.x * blockDim.x) >> 5);
  const int wave = (int)((blockIdx.x * blockDim.x + threadIdx.x) >> 5);

  for (int tile = wave; tile < numTiles; tile += wavesTotal) {
    int p = tile * 16 + row;
    const int pc = (p < N) ? p : (N - 1);   // N % 16 == 0 in practice; keep all lanes live
    const float* xp = x + 3 * (size_t)pc;
    const float x0 = xp[0];
    const float x1 = xp[1];
    const float x2 = xp[2];

    // speculative prefetch of the next tile this wave will touch
    {
      long np = (long)(tile + wavesTotal) * 16 + row;
      if (np < N) __builtin_prefetch(x + 3 * np, 0, 0);
    }

    // B-matrix fragments (4x16 f32): row K striped across lanes (N = lane&15);
    // lanes 0-15 carry K=0,1 ; lanes 16-31 carry K=2,3 (mirrors A layout).
    v2f b1, b2;
    if (half == 0) {
      b1 = (v2f){x0 * x0, x1 * x1};
      b2 = (v2f){x1, x2};
    } else {
      b1 = (v2f){x2 * x2, x0};
      b2 = (v2f){1.0f, 0.0f};
    }

    v8f acc = {};
    acc = __builtin_amdgcn_wmma_f32_16x16x4_f32(false, a1, false, b1,
                                                (short)0, acc, false, false);
    acc = __builtin_amdgcn_wmma_f32_16x16x4_f32(false, a2, false, b2,
                                                (short)0, acc, false, false);

    // D layout: lane holds comps (half*8 .. half*8+7) of point p in acc[0..7];
    // partner lane (xor 16) holds the other 8 comps of the same point.
    float mx = acc[0];
#pragma unroll
    for (int i = 1; i < 8; ++i) mx = fmaxf(mx, acc[i]);
    const float omx = __shfl_xor(mx, 16, 32);
    const float M = fmaxf(mx, omx);
    float s = 0.0f;
#pragma unroll
    for (int i = 0; i < 8; ++i) s += __expf(acc[i] - M);
    const float os = __shfl_xor(s, 16, 32);
    const float r = M + __logf(s + os);

    if (half == 0 && p < N) out[p] = r;
  }
}

extern "C" void kernel_launch(void* const* d_in, const int* in_sizes, int n_in,
                              void* d_out, int out_size, void* d_ws, size_t ws_size,
                              hipStream_t stream) {
  const float* x     = (const float*)d_in[0];
  const float* means = (const float*)d_in[1];
  const float* cov   = (const float*)d_in[2];
  const float* w     = (const float*)d_in[3];
  float* out = (float*)d_out;

  const int N = in_sizes[0] / 3;
  const int numTiles = (N + 15) / 16;

  int blocks = 2048;                       // persistent grid: 16384 waves
  const int wavesPerBlock = 256 / 32;
  if (numTiles < blocks * wavesPerBlock)
    blocks = (numTiles + wavesPerBlock - 1) / wavesPerBlock;
  if (blocks < 1) blocks = 1;

  gmm_logsumexp_wmma<<<blocks, 256, 0, stream>>>(x, means, cov, w, out,
                                                 numTiles, N);
}